// AbstractRelu_70832600645724
// MI455X (gfx1250) — compile-verified
//
#include <hip/hip_runtime.h>

// AbstractRelu (DeepPoly ReLU bounds), elementwise over N=2^24 floats.
// Memory-bound: 384 MiB traffic @ 23.3 TB/s => ~17 us floor. Strategy:
// b128 vector loads/stores with non-temporal (TH=NT) cache policy, one
// float4 chunk per lane, flat launch for maximal memory-level parallelism.

typedef float float4v __attribute__((ext_vector_type(4)));

__device__ __forceinline__ void relu_bounds(float x, float l, float h,
                                            float& xo, float& lo, float& ho) {
    const bool crossing = (l < 0.0f) & (h > 0.0f);
    const bool nonpos   = (h <= 0.0f);

    // safe denom: (h - l) on the crossing branch, 1.0 elsewhere (exact recip)
    const float safe  = crossing ? (h - l) : 1.0f;
    const float rden  = 1.0f / safe;            // IEEE div, cheap vs HBM
    const float slope = h * rden;               // deepPoly UB slope
    const float intc  = -(l * h) * rden;        // deepPoly UB intercept
    const float lamda = (l * l > h * h) ? 0.0f : 1.0f;

    const float w_high = crossing ? slope : (nonpos ? 0.0f : 1.0f);
    const float b_high = crossing ? intc  : 0.0f;
    const float w_low  = crossing ? lamda : (nonpos ? 0.0f : 1.0f);

    xo = fmaxf(x, 0.0f);
    lo = w_low * l;
    ho = fmaf(w_high, h, b_high);
}

__global__ __launch_bounds__(256) void abstract_relu_vec4(
    const float* __restrict__ x,
    const float* __restrict__ low,
    const float* __restrict__ high,
    float* __restrict__ x_out,
    float* __restrict__ low_out,
    float* __restrict__ high_out,
    int n4) {
    const int i = blockIdx.x * blockDim.x + threadIdx.x;
    if (i >= n4) return;

    // 128-bit non-temporal loads: stream-once data, keep it out of L2
    const float4v xv = __builtin_nontemporal_load((const float4v*)x    + i);
    const float4v lv = __builtin_nontemporal_load((const float4v*)low  + i);
    const float4v hv = __builtin_nontemporal_load((const float4v*)high + i);

    float4v xo, lo, ho;
#pragma unroll
    for (int k = 0; k < 4; ++k) {
        float a, b, c;
        relu_bounds(xv[k], lv[k], hv[k], a, b, c);
        xo[k] = a; lo[k] = b; ho[k] = c;
    }

    __builtin_nontemporal_store(xo, (float4v*)x_out    + i);
    __builtin_nontemporal_store(lo, (float4v*)low_out  + i);
    __builtin_nontemporal_store(ho, (float4v*)high_out + i);
}

// Scalar tail for n not divisible by 4 (not hit for N=2^24, kept for safety).
__global__ __launch_bounds__(256) void abstract_relu_tail(
    const float* __restrict__ x,
    const float* __restrict__ low,
    const float* __restrict__ high,
    float* __restrict__ x_out,
    float* __restrict__ low_out,
    float* __restrict__ high_out,
    int base, int n) {
    const int i = base + blockIdx.x * blockDim.x + threadIdx.x;
    if (i >= n) return;
    float a, b, c;
    relu_bounds(x[i], low[i], high[i], a, b, c);
    x_out[i]    = a;
    low_out[i]  = b;
    high_out[i] = c;
}

extern "C" void kernel_launch(void* const* d_in, const int* in_sizes, int n_in,
                              void* d_out, int out_size, void* d_ws, size_t ws_size,
                              hipStream_t stream) {
    const float* x    = (const float*)d_in[0];
    const float* low  = (const float*)d_in[1];
    const float* high = (const float*)d_in[2];
    const int n = in_sizes[0];

    float* x_out    = (float*)d_out;         // outputs concatenated flat
    float* low_out  = x_out + n;
    float* high_out = low_out + n;

    const int n4 = n >> 2;
    if (n4 > 0) {
        const int threads = 256;
        const int blocks  = (n4 + threads - 1) / threads;
        abstract_relu_vec4<<<blocks, threads, 0, stream>>>(
            x, low, high, x_out, low_out, high_out, n4);
    }
    const int rem = n - (n4 << 2);
    if (rem > 0) {
        abstract_relu_tail<<<1, 256, 0, stream>>>(
            x, low, high, x_out, low_out, high_out, n4 << 2, n);
    }
}